// yoloLoss_4896262717443
// MI455X (gfx1250) — compile-verified
//
#include <hip/hip_runtime.h>
#include <math.h>

#define S 14
#define CH 30
#define TPB 256
#define STEPF (1.0f / 14.0f)

typedef float f4  __attribute__((ext_vector_type(4)));
typedef int   v4i __attribute__((ext_vector_type(4)));

// CDNA5 async global->LDS path (gfx1250), guarded so either toolchain compiles.
#if defined(__HIP_DEVICE_COMPILE__) && defined(__gfx1250__) && \
    __has_builtin(__builtin_amdgcn_global_load_async_to_lds_b128)
#define USE_ASYNC 1
typedef __attribute__((address_space(1))) v4i gv4i;   // global int4
typedef __attribute__((address_space(3))) v4i lv4i;   // LDS int4
#endif

static __device__ __forceinline__ float smooth_l1(float p, float t) {
    float d  = p - t;
    float ad = fabsf(d);
    return (ad < 1.0f) ? 0.5f * d * d : (ad - 0.5f);
}

// IOU of converted boxes; pb/tb point at raw [x,y,w,h] (LDS), fi/fj = cell coords.
static __device__ __forceinline__ float iou_conv(const float* pb, const float* tb,
                                                 float fi, float fj) {
    float x1 = fmaxf((pb[0] + fi) * STEPF - pb[2] * 0.5f, 0.0f);
    float y1 = fmaxf((pb[1] + fj) * STEPF - pb[3] * 0.5f, 0.0f);
    float w1 = fmaxf(pb[2], 0.0f);
    float h1 = fmaxf(pb[3], 0.0f);
    float x2 = fmaxf((tb[0] + fi) * STEPF - tb[2] * 0.5f, 0.0f);
    float y2 = fmaxf((tb[1] + fj) * STEPF - tb[3] * 0.5f, 0.0f);
    float w2 = fmaxf(tb[2], 0.0f);
    float h2 = fmaxf(tb[3], 0.0f);
    float iw = fmaxf(w1 + w2 - (fmaxf(x1 + w1, x2 + w2) - fminf(x1, x2)), 0.0f);
    float ih = fmaxf(h1 + h2 - (fmaxf(y1 + h1, y2 + h2) - fminf(y1, y2)), 0.0f);
    float inter = iw * ih;
    float uni   = w1 * h1 + w2 * h2 - inter;
    return inter / (uni + 1e-6f);
}

__global__ __launch_bounds__(TPB) void yolo_loss_main(
        const float* __restrict__ pred, const float* __restrict__ tgt,
        float* __restrict__ partials, int ncells) {
    __shared__ __align__(16) float sm_p[TPB * CH];   // 30720 B
    __shared__ __align__(16) float sm_t[TPB * CH];   // 30720 B
    __shared__ float red[TPB];

    const int tid    = threadIdx.x;
    const int cell0  = blockIdx.x * TPB;
    const int elems  = min(TPB, ncells - cell0);
    const int nfloat = elems * CH;
    const int nvec   = nfloat >> 2;                   // chunk base is 16B aligned
    const float* gp = pred + (size_t)cell0 * CH;
    const float* gt = tgt  + (size_t)cell0 * CH;

#if USE_ASYNC
    // Per-lane 16B async DMA copies, tracked by ASYNCcnt (no VGPR staging).
    for (int v = tid; v < nvec; v += TPB) {
        const v4i* s0 = (const v4i*)(gp + 4 * v);
        const v4i* s1 = (const v4i*)(gt + 4 * v);
        v4i*       d0 = (v4i*)(sm_p + 4 * v);
        v4i*       d1 = (v4i*)(sm_t + 4 * v);
        __builtin_amdgcn_global_load_async_to_lds_b128((gv4i*)s0, (lv4i*)d0, 0, 0);
        __builtin_amdgcn_global_load_async_to_lds_b128((gv4i*)s1, (lv4i*)d1, 0, 0);
    }
    for (int v = (nvec << 2) + tid; v < nfloat; v += TPB) {  // scalar tail (none for full blocks)
        sm_p[v] = gp[v];
        sm_t[v] = gt[v];
    }
#if __has_builtin(__builtin_amdgcn_s_wait_asynccnt)
    __builtin_amdgcn_s_wait_asynccnt(0);
#else
    asm volatile("s_wait_asynccnt 0x0" ::: "memory");
#endif
#else
    // Fallback: non-temporal 128-bit loads (single-pass stream; don't pollute L2).
    for (int v = tid; v < nvec; v += TPB) {
        f4 a = __builtin_nontemporal_load((const f4*)gp + v);
        f4 b = __builtin_nontemporal_load((const f4*)gt + v);
        *(f4*)(sm_p + 4 * v) = a;
        *(f4*)(sm_t + 4 * v) = b;
    }
    for (int v = (nvec << 2) + tid; v < nfloat; v += TPB) {
        sm_p[v] = gp[v];
        sm_t[v] = gt[v];
    }
#endif
    __syncthreads();

    float partial = 0.0f;
    const int cell = cell0 + tid;
    if (tid < elems) {
        const float* pp = sm_p + tid * CH;
        const float* tt = sm_t + tid * CH;
        const int   i  = cell % S;
        const int   j  = (cell / S) % S;
        const float fi = (float)i;
        const float fj = (float)j;

        float iou0 = iou_conv(pp + 0, tt + 0, fi, fj);
        float iou1 = iou_conv(pp + 5, tt + 5, fi, fj);
        const int  maxi = (iou1 > iou0) ? 1 : 0;   // first-occurrence argmax
        const bool sig  = tt[9] > 0.0f;

        float obj_l = 0.0f, noobj_l = 0.0f, xy_l = 0.0f, wh_l = 0.0f, cls_l = 0.0f;
#pragma unroll
        for (int k = 0; k < 2; ++k) {
            bool obj = tt[5 * k + 4] > 0.0f;
            if (sig) obj = obj && (k == maxi);
            float dc  = pp[5 * k + 4] - tt[5 * k + 4];
            float csq = dc * dc;
            if (obj) {
                obj_l += csq;
                xy_l  += smooth_l1(pp[5 * k + 0], tt[5 * k + 0]) +
                         smooth_l1(pp[5 * k + 1], tt[5 * k + 1]);
                wh_l  += smooth_l1(sqrtf(fmaxf(pp[5 * k + 2], 1e-6f)),
                                   sqrtf(fmaxf(tt[5 * k + 2], 1e-6f))) +
                         smooth_l1(sqrtf(fmaxf(pp[5 * k + 3], 1e-6f)),
                                   sqrtf(fmaxf(tt[5 * k + 3], 1e-6f)));
            } else {
                noobj_l += csq;
            }
        }

        if (sig) {
            float m = pp[10];
#pragma unroll
            for (int c = 1; c < 20; ++c) m = fmaxf(m, pp[10 + c]);
            float s = 0.0f;
#pragma unroll
            for (int c = 0; c < 20; ++c) s += expf(pp[10 + c] - m);
            int   ti = 0;
            float tm = tt[10];
#pragma unroll
            for (int c = 1; c < 20; ++c) {
                if (tt[10 + c] > tm) { tm = tt[10 + c]; ti = c; }  // first max
            }
            cls_l = -((pp[10 + ti] - m) - logf(s));
        }

        partial = 3.0f * (xy_l + wh_l) + 1.0f * obj_l + 0.3f * noobj_l + 1.5f * cls_l;
    }

    // Deterministic fixed-order block reduction.
    red[tid] = partial;
    __syncthreads();
#pragma unroll
    for (int s = TPB / 2; s > 0; s >>= 1) {
        if (tid < s) red[tid] += red[tid + s];
        __syncthreads();
    }
    if (tid == 0) partials[blockIdx.x] = red[0];
}

__global__ __launch_bounds__(TPB) void yolo_loss_reduce(
        const float* __restrict__ partials, float* __restrict__ out,
        int n, float inv_batch) {
    __shared__ float red[TPB];
    float acc = 0.0f;
    for (int v = threadIdx.x; v < n; v += TPB) acc += partials[v];  // fixed order
    red[threadIdx.x] = acc;
    __syncthreads();
#pragma unroll
    for (int s = TPB / 2; s > 0; s >>= 1) {
        if (threadIdx.x < s) red[threadIdx.x] += red[threadIdx.x + s];
        __syncthreads();
    }
    if (threadIdx.x == 0) out[0] = red[0] * inv_batch;
}

extern "C" void kernel_launch(void* const* d_in, const int* in_sizes, int n_in,
                              void* d_out, int out_size, void* d_ws, size_t ws_size,
                              hipStream_t stream) {
    const float* pred = (const float*)d_in[0];
    const float* tgt  = (const float*)d_in[1];
    const int n0      = in_sizes[0];          // batch*S*S*30
    const int ncells  = n0 / CH;              // 802816
    const int batch   = ncells / (S * S);     // 4096
    const int nblocks = (ncells + TPB - 1) / TPB;  // 3136 exactly

    float* partials = (float*)d_ws;           // nblocks floats of scratch
    yolo_loss_main<<<nblocks, TPB, 0, stream>>>(pred, tgt, partials, ncells);
    yolo_loss_reduce<<<1, TPB, 0, stream>>>(partials, (float*)d_out, nblocks,
                                            1.0f / (float)batch);
}